// MixerModel_50259707298213
// MI455X (gfx1250) — compile-verified
//
#include <hip/hip_runtime.h>
#include <cmath>

#define D_MODEL 1024
#define N_LAYER 2
#define D_INNER 2048
#define D_STATE 16
#define D_CONV 4
#define DT_RANK 64
#define BATCH 2
#define SEQLEN 2048
#define BL (BATCH * SEQLEN)
#define XDBL_W (DT_RANK + 2 * D_STATE) /* 96 */
#define EPS 1e-5f
#define SCAN_CHUNK 64

typedef float v2f __attribute__((ext_vector_type(2)));
typedef float v8f __attribute__((ext_vector_type(8)));

__device__ __forceinline__ float silu_f(float x) {
    return x * (1.0f / (1.0f + __expf(-x)));
}
__device__ __forceinline__ float softplus_f(float x) {
    return x > 20.0f ? x : log1pf(__expf(x));
}

// ---------------------------------------------------------------------------
// Fused (residual += hidden) + LayerNorm over rows of D_MODEL.
// One block (256 threads = 8 waves) per row. hidden/normed_out may alias
// (read happens before the barrier, write after) -> no __restrict__ on them.
// ---------------------------------------------------------------------------
__global__ __launch_bounds__(256)
void add_norm_kernel(const float* hidden, const float* __restrict__ residual_in,
                     float* __restrict__ residual_out, float* normed_out,
                     const float* __restrict__ w, const float* __restrict__ b) {
    const size_t row = blockIdx.x;
    const float* hp = hidden + row * D_MODEL;
    const float* rp = residual_in ? residual_in + row * D_MODEL : nullptr;
    float* rop = residual_out + row * D_MODEL;
    float* nop = normed_out + row * D_MODEL;

    float x[D_MODEL / 256];
    float s = 0.0f, q = 0.0f;
#pragma unroll
    for (int j = 0; j < D_MODEL / 256; ++j) {
        int c = threadIdx.x + j * 256;
        float v = hp[c] + (rp ? rp[c] : 0.0f);
        x[j] = v;
        rop[c] = v;
        s += v;
        q += v * v;
    }
    // wave32 reduction then cross-wave via LDS
#pragma unroll
    for (int o = 16; o > 0; o >>= 1) {
        s += __shfl_xor(s, o, 32);
        q += __shfl_xor(q, o, 32);
    }
    __shared__ float sw[8], sq[8];
    int wid = threadIdx.x >> 5;
    if ((threadIdx.x & 31) == 0) { sw[wid] = s; sq[wid] = q; }
    __syncthreads();
    float ts = 0.0f, tq = 0.0f;
#pragma unroll
    for (int i = 0; i < 8; ++i) { ts += sw[i]; tq += sq[i]; }
    const float mu = ts * (1.0f / D_MODEL);
    const float var = tq * (1.0f / D_MODEL) - mu * mu;
    const float rstd = rsqrtf(var + EPS);
#pragma unroll
    for (int j = 0; j < D_MODEL / 256; ++j) {
        int c = threadIdx.x + j * 256;
        nop[c] = (x[j] - mu) * rstd * w[c] + b[c];
    }
}

// ---------------------------------------------------------------------------
// fp32 WMMA GEMM: C = f(A[M,K] * W[N,K]^T + bias), register-blocked.
// Each wave owns a (16*MT) x (16*NT) output tile: MT*NT v8f accumulators.
// Per K-step (K += 4): MT + NT b64 fragment loads feed MT*NT
// V_WMMA_F32_16X16X4_F32 ops -> 128 B per WMMA at MT=NT=4 (16 flop/byte).
// Fragment mapping (ISA 7.12.2): lanes 0-15 carry K={k,k+1} of row/col
// `idx`, lanes 16-31 carry K={k+2,k+3}; C/D vgpr i -> M = i + 8*half.
// MODE 0: store; MODE 1: softplus(x + bias[n]) (dt_proj).
// ---------------------------------------------------------------------------
template <int MT, int NT, int MODE>
__global__ __launch_bounds__(32)
void wmma_gemm_kernel(const float* __restrict__ A, const float* __restrict__ W,
                      const float* __restrict__ bias, float* __restrict__ C,
                      int K, int lda, int ldb, int ldc) {
    const int lane = threadIdx.x & 31;
    const int half = lane >> 4;
    const int idx = lane & 15;
    const int row0 = blockIdx.y * (16 * MT);
    const int col0 = blockIdx.x * (16 * NT);

    const float* ap[MT];
    const float* bp[NT];
#pragma unroll
    for (int i = 0; i < MT; ++i)
        ap[i] = A + (size_t)(row0 + 16 * i + idx) * lda + 2 * half;
#pragma unroll
    for (int j = 0; j < NT; ++j)
        bp[j] = W + (size_t)(col0 + 16 * j + idx) * ldb + 2 * half;

    const v8f zero = {};
    v8f acc[MT][NT];
#pragma unroll
    for (int i = 0; i < MT; ++i)
#pragma unroll
        for (int j = 0; j < NT; ++j) acc[i][j] = zero;

    for (int k = 0; k < K; k += 4) {
        v2f a[MT], b[NT];
#pragma unroll
        for (int i = 0; i < MT; ++i) a[i] = *(const v2f*)(ap[i] + k);
#pragma unroll
        for (int j = 0; j < NT; ++j) b[j] = *(const v2f*)(bp[j] + k);
#pragma unroll
        for (int i = 0; i < MT; ++i)
#pragma unroll
            for (int j = 0; j < NT; ++j)
                acc[i][j] = __builtin_amdgcn_wmma_f32_16x16x4_f32(
                    false, a[i], false, b[j], (short)0, acc[i][j], false, false);
    }

#pragma unroll
    for (int j = 0; j < NT; ++j) {
        const int cc = col0 + 16 * j + idx;
        const float bv = (MODE == 1) ? bias[cc] : 0.0f;
#pragma unroll
        for (int i = 0; i < MT; ++i) {
#pragma unroll
            for (int t = 0; t < 8; ++t) {
                int rr = row0 + 16 * i + t + half * 8;
                float v = acc[i][j][t] + bv;
                if (MODE == 1) v = softplus_f(v);
                C[(size_t)rr * ldc + cc] = v;
            }
        }
    }
}

// ---------------------------------------------------------------------------
// Depthwise causal conv1d (width 4) + SiLU. xi = xz[:, :, 0:D_INNER].
// ---------------------------------------------------------------------------
__global__ __launch_bounds__(256)
void conv_silu_kernel(const float* __restrict__ xz, const float* __restrict__ cw,
                      const float* __restrict__ cb, float* __restrict__ xc) {
    const size_t idx = (size_t)blockIdx.x * blockDim.x + threadIdx.x;
    if (idx >= (size_t)BL * D_INNER) return;
    const int d = (int)(idx % D_INNER);
    const size_t r = idx / D_INNER;            // combined (b, l) row
    const int l = (int)(r % SEQLEN);
    const size_t rowbase = r - l;              // b * SEQLEN

    float acc = cb[d];
#pragma unroll
    for (int k = 0; k < D_CONV; ++k) {
        int ls = l - (D_CONV - 1) + k;
        if (ls >= 0)
            acc += xz[(rowbase + ls) * (2 * D_INNER) + d] * cw[d * D_CONV + k];
    }
    xc[idx] = silu_f(acc);
}

// ---------------------------------------------------------------------------
// Selective scan. One lane per (batch, channel): h[16] lives in VGPRs,
// loops the 2048 sequential steps. The shared B/C rows are staged into LDS
// with CDNA5 async DMA (GLOBAL_LOAD_ASYNC_TO_LDS_B128, L2 -> LDS without
// touching VGPRs), synchronized via s_wait_asynccnt + barrier. Output is
// fused with D-skip and SiLU(z) gate. y may alias dt (same-thread
// read-before-write at identical index).
// ---------------------------------------------------------------------------
__global__ __launch_bounds__(256)
void scan_kernel(const float* __restrict__ dt, const float* __restrict__ xc,
                 const float* __restrict__ xdbl, const float* __restrict__ xz,
                 const float* __restrict__ A_log, const float* __restrict__ Dsk,
                 float* __restrict__ y) {
    __shared__ float sBC[SCAN_CHUNK * 2 * D_STATE];

    const int groups = D_INNER / 256;              // 8
    const int b = blockIdx.x / groups;
    const int g = blockIdx.x % groups;
    const int d = g * 256 + threadIdx.x;

    float A[D_STATE], h[D_STATE];
#pragma unroll
    for (int s = 0; s < D_STATE; ++s) {
        A[s] = -__expf(A_log[d * D_STATE + s]);
        h[s] = 0.0f;
    }
    const float Dv = Dsk[d];
    const size_t rowbase = (size_t)b * SEQLEN;

    for (int l0 = 0; l0 < SEQLEN; l0 += SCAN_CHUNK) {
        __syncthreads();
        // Async DMA stage: 64 steps x 32 floats, 16 B per lane per issue.
        // Global addr (row*96 + 64 + 4k)*4 B and LDS addr t*16 B are both
        // 16-byte aligned.
        for (int t = threadIdx.x; t < SCAN_CHUNK * 8; t += 256) {
            const int ll = t >> 3;
            const int s4 = (t & 7) * 4;
            const float* gsrc =
                &xdbl[(rowbase + l0 + ll) * XDBL_W + DT_RANK + s4];
            const unsigned lds_off =
                (unsigned)(size_t)(&sBC[ll * 2 * D_STATE + s4]);
            asm volatile("global_load_async_to_lds_b128 %0, %1, off"
                         :: "v"(lds_off), "v"(gsrc) : "memory");
        }
        asm volatile("s_wait_asynccnt 0x0" ::: "memory");
        __syncthreads();

        for (int li = 0; li < SCAN_CHUNK; ++li) {
            const size_t r = rowbase + l0 + li;
            const float dtv = dt[r * D_INNER + d];
            const float xv = xc[r * D_INNER + d];
            const float zv = xz[r * (2 * D_INNER) + D_INNER + d];
            const float* Bc = &sBC[li * 2 * D_STATE];
            const float* Cc = Bc + D_STATE;
            const float dx = dtv * xv;
            float yv = 0.0f;
#pragma unroll
            for (int s = 0; s < D_STATE; ++s) {
                h[s] = __expf(dtv * A[s]) * h[s] + dx * Bc[s];
                yv += h[s] * Cc[s];
            }
            yv += xv * Dv;
            y[r * D_INNER + d] = yv * silu_f(zv);
        }
    }
}

// ---------------------------------------------------------------------------
extern "C" void kernel_launch(void* const* d_in, const int* in_sizes, int n_in,
                              void* d_out, int out_size, void* d_ws, size_t ws_size,
                              hipStream_t stream) {
    const float* hs     = (const float*)d_in[0];
    const float* norm_w = (const float*)d_in[1];
    const float* norm_b = (const float*)d_in[2];
    const float* in_w   = (const float*)d_in[3];
    const float* conv_w = (const float*)d_in[4];
    const float* conv_b = (const float*)d_in[5];
    const float* x_w    = (const float*)d_in[6];
    const float* dt_w   = (const float*)d_in[7];
    const float* dt_b   = (const float*)d_in[8];
    const float* A_log  = (const float*)d_in[9];
    const float* D_sk   = (const float*)d_in[10];
    const float* out_w  = (const float*)d_in[11];
    const float* nf_w   = (const float*)d_in[12];
    const float* nf_b   = (const float*)d_in[13];
    float* out = (float*)d_out;

    // Workspace layout (~162 MB). Every buffer is fully written before read.
    char* base = (char*)d_ws;
    size_t off = 0;
    auto alloc = [&](size_t elems) {
        float* p = (float*)(base + off);
        off += elems * sizeof(float);
        return p;
    };
    float* resid  = alloc((size_t)BL * D_MODEL);       // persistent residual
    float* normed = alloc((size_t)BL * D_MODEL);       // LN out; reused as out_proj dst
    float* xz     = alloc((size_t)BL * 2 * D_INNER);   // in_proj out (xi | z)
    float* xc     = alloc((size_t)BL * D_INNER);       // conv+silu out
    float* xdbl   = alloc((size_t)BL * XDBL_W);        // x_proj out
    float* dtbuf  = alloc((size_t)BL * D_INNER);       // dt; scan rewrites as y

    const float* hidden = hs;
    const float* res_in = nullptr;

    for (int i = 0; i < N_LAYER; ++i) {
        add_norm_kernel<<<BL, 256, 0, stream>>>(
            hidden, res_in, resid, normed,
            norm_w + (size_t)i * D_MODEL, norm_b + (size_t)i * D_MODEL);

        // xz = normed @ in_w^T   [4096 x 4096], K=1024, 64x64 tiles
        wmma_gemm_kernel<4, 4, 0>
            <<<dim3(2 * D_INNER / 64, BL / 64), 32, 0, stream>>>(
            normed, in_w + (size_t)i * 2 * D_INNER * D_MODEL, nullptr, xz,
            D_MODEL, D_MODEL, D_MODEL, 2 * D_INNER);

        conv_silu_kernel<<<(BL * D_INNER) / 256, 256, 0, stream>>>(
            xz, conv_w + (size_t)i * D_INNER * D_CONV, conv_b + (size_t)i * D_INNER, xc);

        // x_dbl = xc @ x_w^T   [4096 x 96], K=2048, 64x32 tiles
        wmma_gemm_kernel<4, 2, 0>
            <<<dim3(XDBL_W / 32, BL / 64), 32, 0, stream>>>(
            xc, x_w + (size_t)i * XDBL_W * D_INNER, nullptr, xdbl,
            D_INNER, D_INNER, D_INNER, XDBL_W);

        // dt = softplus(x_dbl[:, :64] @ dt_w^T + dt_b)  [4096 x 2048], K=64, lda=96
        wmma_gemm_kernel<4, 4, 1>
            <<<dim3(D_INNER / 64, BL / 64), 32, 0, stream>>>(
            xdbl, dt_w + (size_t)i * D_INNER * DT_RANK, dt_b + (size_t)i * D_INNER,
            dtbuf, DT_RANK, XDBL_W, DT_RANK, D_INNER);

        scan_kernel<<<(BATCH * D_INNER) / 256, 256, 0, stream>>>(
            dtbuf, xc, xdbl, xz,
            A_log + (size_t)i * D_INNER * D_STATE, D_sk + (size_t)i * D_INNER, dtbuf);

        // hidden = y @ out_w^T   [4096 x 1024], K=2048  (into `normed`, now free)
        wmma_gemm_kernel<4, 4, 0>
            <<<dim3(D_MODEL / 64, BL / 64), 32, 0, stream>>>(
            dtbuf, out_w + (size_t)i * D_MODEL * D_INNER, nullptr, normed,
            D_INNER, D_INNER, D_INNER, D_MODEL);

        hidden = normed;
        res_in = resid;
    }

    // final: out = LN(hidden + residual; normf)
    add_norm_kernel<<<BL, 256, 0, stream>>>(hidden, res_in, resid, out, nf_w, nf_b);
}